// GCNModel_13477607375482
// MI455X (gfx1250) — compile-verified
//
#include <hip/hip_runtime.h>
#include <math.h>

// GCN forward: N=100000 nodes, E=1.6M edges, F_IN=64, H=64, C=40.
// fp32 throughout (matches reference numerics) using V_WMMA_F32_16X16X4_F32
// for both GEMMs; scatter-add via global fp32 atomics.

typedef __attribute__((ext_vector_type(2))) float v2f;
typedef __attribute__((ext_vector_type(8))) float v8f;

__device__ __forceinline__ void atomAddF32(float* p, float v) {
    __hip_atomic_fetch_add(p, v, __ATOMIC_RELAXED, __HIP_MEMORY_SCOPE_AGENT);
}

// ---------------- degree / normalization ----------------

__global__ void k_deg_init(float* __restrict__ deg, int n) {
    int i = blockIdx.x * blockDim.x + threadIdx.x;
    if (i < n) deg[i] = 1.0f;  // self-loop weight
}

__global__ void k_deg_scatter(const int* __restrict__ dst, const float* __restrict__ w,
                              float* __restrict__ deg, int E) {
    int e = blockIdx.x * blockDim.x + threadIdx.x;
    if (e < E) atomAddF32(&deg[dst[e]], w[e]);
}

__global__ void k_dinv(float* __restrict__ deg, int n) {
    int i = blockIdx.x * blockDim.x + threadIdx.x;
    if (i < n) {
        float d = deg[i];
        deg[i] = (d > 0.0f) ? rsqrtf(d) : 0.0f;
    }
}

__global__ void k_norm(const int* __restrict__ src, const int* __restrict__ dst,
                       const float* __restrict__ w, const float* __restrict__ dinv,
                       float* __restrict__ nrm, int E) {
    int e = blockIdx.x * blockDim.x + threadIdx.x;
    if (e < E) nrm[e] = dinv[src[e]] * w[e] * dinv[dst[e]];
}

// ---------------- GEMM1: [N,64] @ [64,64], fp32 WMMA ----------------
// One wave per 16x16 output tile; K in steps of 4 (16 WMMA ops).

__global__ void k_gemm1(const float* __restrict__ X, const float* __restrict__ W,
                        float* __restrict__ O, int mtiles) {
    int wave = threadIdx.x >> 5;
    int tile = blockIdx.x * (blockDim.x >> 5) + wave;
    if (tile >= mtiles * 4) return;
    int mt = tile >> 2, nt = tile & 3;
    int lane = threadIdx.x & 31;
    int lo = lane & 15, hi = lane >> 4;

    const float* arow = X + (size_t)(mt * 16 + lo) * 64;  // A: row M=lo
    int col = nt * 16 + lo;                                // B/C/D: col N=lo
    v8f c = {};
#pragma unroll
    for (int k = 0; k < 64; k += 4) {
        int ka = k + 2 * hi;  // lanes 0-15: K=k,k+1 ; lanes 16-31: K=k+2,k+3
        v2f a = { arow[ka], arow[ka + 1] };
        v2f b = { W[(size_t)ka * 64 + col], W[(size_t)(ka + 1) * 64 + col] };
        c = __builtin_amdgcn_wmma_f32_16x16x4_f32(false, a, false, b,
                                                  (short)0, c, false, false);
    }
    float* orow = O + (size_t)(mt * 16 + 8 * hi) * 64 + col;
#pragma unroll
    for (int r = 0; r < 8; ++r) orow[(size_t)r * 64] = c[r];  // M = r + 8*hi
}

// ---------------- GEMM2: relu([N,64]) @ [64,40], fp32 WMMA ----------------
// 3 n-tiles cover 48 cols; cols >=40 are zero-padded / store-guarded.

__global__ void k_gemm2(const float* __restrict__ Hin, const float* __restrict__ W2,
                        float* __restrict__ O, int mtiles) {
    int wave = threadIdx.x >> 5;
    int tile = blockIdx.x * (blockDim.x >> 5) + wave;
    if (tile >= mtiles * 3) return;
    int mt = tile / 3, nt = tile - mt * 3;
    int lane = threadIdx.x & 31;
    int lo = lane & 15, hi = lane >> 4;

    const float* arow = Hin + (size_t)(mt * 16 + lo) * 64;
    int col = nt * 16 + lo;
    bool cok = col < 40;
    v8f c = {};
#pragma unroll
    for (int k = 0; k < 64; k += 4) {
        int ka = k + 2 * hi;
        v2f a = { fmaxf(arow[ka], 0.0f), fmaxf(arow[ka + 1], 0.0f) };  // fused ReLU
        v2f b = { cok ? W2[(size_t)ka * 40 + col] : 0.0f,
                  cok ? W2[(size_t)(ka + 1) * 40 + col] : 0.0f };
        c = __builtin_amdgcn_wmma_f32_16x16x4_f32(false, a, false, b,
                                                  (short)0, c, false, false);
    }
    if (cok) {
        float* orow = O + (size_t)(mt * 16 + 8 * hi) * 40 + col;
#pragma unroll
        for (int r = 0; r < 8; ++r) orow[(size_t)r * 40] = c[r];
    }
}

// ---------------- self-loop message + bias init ----------------

__global__ void k_self1(const float* __restrict__ h, const float* __restrict__ dinv,
                        const float* __restrict__ bias, float* __restrict__ out, int n) {
    int t = blockIdx.x * blockDim.x + threadIdx.x;
    if (t < n * 64) {
        int i = t >> 6, f = t & 63;
        float di = dinv[i];
        out[t] = bias[f] + di * di * h[t];
    }
}

__global__ void k_self2(const float* __restrict__ h, const float* __restrict__ dinv,
                        const float* __restrict__ bias, float* __restrict__ out, int n) {
    int t = blockIdx.x * blockDim.x + threadIdx.x;
    if (t < n * 64) {
        int i = t >> 6, f = t & 63;
        if (f < 40) {
            float di = dinv[i];
            out[(size_t)i * 40 + f] = bias[f] + di * di * h[(size_t)i * 40 + f];
        }
    }
}

// ---------------- edge scatter (atomic fadd) ----------------

__global__ void k_scat1(const int* __restrict__ src, const int* __restrict__ dst,
                        const float* __restrict__ nrm, const float* __restrict__ h,
                        float* __restrict__ out, int E) {
    long long t = (long long)blockIdx.x * blockDim.x + threadIdx.x;
    if (t >= (long long)E * 64) return;
    int e = (int)(t >> 6), f = (int)(t & 63);
    int s = src[e], d = dst[e];
    atomAddF32(&out[(size_t)d * 64 + f], nrm[e] * h[(size_t)s * 64 + f]);
}

__global__ void k_scat2(const int* __restrict__ src, const int* __restrict__ dst,
                        const float* __restrict__ nrm, const float* __restrict__ h,
                        float* __restrict__ out, int E) {
    long long t = (long long)blockIdx.x * blockDim.x + threadIdx.x;
    if (t >= (long long)E * 64) return;
    int e = (int)(t >> 6), f = (int)(t & 63);
    if (f < 40) {
        int s = src[e], d = dst[e];
        atomAddF32(&out[(size_t)d * 40 + f], nrm[e] * h[(size_t)s * 40 + f]);
    }
}

// ---------------- log_softmax over 40 classes ----------------

__global__ void k_lsm(const float* __restrict__ h, float* __restrict__ out, int n) {
    int i = blockIdx.x * blockDim.x + threadIdx.x;
    if (i >= n) return;
    float v[40];
    float mx = -3.402823466e38f;
#pragma unroll
    for (int c = 0; c < 40; ++c) { v[c] = h[(size_t)i * 40 + c]; mx = fmaxf(mx, v[c]); }
    float s = 0.0f;
#pragma unroll
    for (int c = 0; c < 40; ++c) s += expf(v[c] - mx);
    float l = mx + logf(s);
#pragma unroll
    for (int c = 0; c < 40; ++c) out[(size_t)i * 40 + c] = v[c] - l;
}

// ---------------- launch ----------------

extern "C" void kernel_launch(void* const* d_in, const int* in_sizes, int n_in,
                              void* d_out, int out_size, void* d_ws, size_t ws_size,
                              hipStream_t stream) {
    const float* x   = (const float*)d_in[0];
    const int*   ei  = (const int*)d_in[1];
    const float* ew  = (const float*)d_in[2];
    const float* W1  = (const float*)d_in[3];
    const float* b1  = (const float*)d_in[4];
    const float* W2  = (const float*)d_in[5];
    const float* b2  = (const float*)d_in[6];
    float* out = (float*)d_out;

    const int N = in_sizes[0] / 64;
    const int E = in_sizes[2];
    const int* src = ei;
    const int* dst = ei + E;

    // Workspace layout (floats): dinv[N] | norm[E] | bufA[N*64] | bufB[N*64]
    float* ws   = (float*)d_ws;
    float* dinv = ws;
    float* nrm  = dinv + N;
    float* bufA = nrm + E;
    float* bufB = bufA + (size_t)N * 64;

    const int BLK = 256;
    const int mtiles = N / 16;  // N=100000 -> 6250 full tiles

    k_deg_init<<<(N + BLK - 1) / BLK, BLK, 0, stream>>>(dinv, N);
    k_deg_scatter<<<(E + BLK - 1) / BLK, BLK, 0, stream>>>(dst, ew, dinv, E);
    k_dinv<<<(N + BLK - 1) / BLK, BLK, 0, stream>>>(dinv, N);
    k_norm<<<(E + BLK - 1) / BLK, BLK, 0, stream>>>(src, dst, ew, dinv, nrm, E);

    // Layer 1: GEMM (WMMA) -> self-loop+bias init -> edge scatter
    k_gemm1<<<(mtiles * 4 + 7) / 8, BLK, 0, stream>>>(x, W1, bufA, mtiles);
    {
        long long t = (long long)N * 64;
        k_self1<<<(unsigned)((t + BLK - 1) / BLK), BLK, 0, stream>>>(bufA, dinv, b1, bufB, N);
    }
    {
        long long t = (long long)E * 64;
        k_scat1<<<(unsigned)((t + BLK - 1) / BLK), BLK, 0, stream>>>(src, dst, nrm, bufA, bufB, E);
    }

    // Layer 2: ReLU fused into GEMM A-loads (WMMA) -> init -> scatter
    k_gemm2<<<(mtiles * 3 + 7) / 8, BLK, 0, stream>>>(bufB, W2, bufA, mtiles);
    {
        long long t = (long long)N * 64;
        k_self2<<<(unsigned)((t + BLK - 1) / BLK), BLK, 0, stream>>>(bufA, dinv, b2, bufB, N);
    }
    {
        long long t = (long long)E * 64;
        k_scat2<<<(unsigned)((t + BLK - 1) / BLK), BLK, 0, stream>>>(src, dst, nrm, bufA, bufB, E);
    }

    k_lsm<<<(N + BLK - 1) / BLK, BLK, 0, stream>>>(bufB, out, N);
}